// PostProcessor_62637803044984
// MI455X (gfx1250) — compile-verified
//
#include <hip/hip_runtime.h>
#include <math.h>

#define B_IMG 2
#define N_BOX 1000
#define C_ALL 81
#define C1 80
#define NCLS (B_IMG * C1)   // 160
#define NPAD 1024
#define NTILE 63            // ceil(1000/16) -> rows/cols 0..1007 (<1024 pad)
#define DETS 100
#define SCORE_TH 0.05f
#define NEG_INF_F (-1e30f)
#define DEAD_F (-3e30f)
#define IMG_W_F 1333.0f
#define IMG_H_F 800.0f
#define XFORM_CLIP 4.135166556742356f  // log(1000/16)

typedef float v2f __attribute__((ext_vector_type(2)));
typedef float v8f __attribute__((ext_vector_type(8)));
typedef unsigned uv4 __attribute__((ext_vector_type(4)));
typedef unsigned uv8 __attribute__((ext_vector_type(8)));

// ---------------- kernel 1: softmax row stats ----------------
__global__ void k_softmax_stats(const float* __restrict__ logits,
                                float* __restrict__ rowmax,
                                float* __restrict__ rowsum) {
  int row = blockIdx.x * blockDim.x + threadIdx.x;
  if (row >= B_IMG * N_BOX) return;
  const float* p = logits + row * C_ALL;
  float m = p[0];
  for (int c = 1; c < C_ALL; ++c) m = fmaxf(m, p[c]);
  float s = 0.f;
  for (int c = 0; c < C_ALL; ++c) s += __expf(p[c] - m);
  rowmax[row] = m;
  rowsum[row] = s;
}

// ---------------- kernel 2: prob + box decode + clip ----------------
__global__ void k_decode(const float* __restrict__ logits,
                         const float* __restrict__ boxreg,
                         const float* __restrict__ props,
                         const float* __restrict__ rowmax,
                         const float* __restrict__ rowsum,
                         float* __restrict__ scores,
                         float* __restrict__ boxes) {
  int t = blockIdx.x * blockDim.x + threadIdx.x;
  if (t >= NCLS * N_BOX) return;
  int n = t % N_BOX;
  int cls = t / N_BOX;          // b*80 + c1
  int b = cls / C1, c1 = cls % C1;
  int row = b * N_BOX + n;
  int c = c1 + 1;               // skip background
  float prob = __expf(logits[row * C_ALL + c] - rowmax[row]) / rowsum[row];

  const float* pb = props + row * 4;
  float w  = pb[2] - pb[0] + 1.f;
  float h  = pb[3] - pb[1] + 1.f;
  float cx = pb[0] + 0.5f * w;
  float cy = pb[1] + 0.5f * h;
  const float* r = boxreg + row * C_ALL * 4 + c * 4;
  float dx = r[0] * 0.1f, dy = r[1] * 0.1f;
  float dw = fminf(r[2] * 0.2f, XFORM_CLIP);
  float dh = fminf(r[3] * 0.2f, XFORM_CLIP);
  float pcx = dx * w + cx, pcy = dy * h + cy;
  float pw = __expf(dw) * w, ph = __expf(dh) * h;
  float x1 = pcx - 0.5f * pw, y1 = pcy - 0.5f * ph;
  float x2 = pcx + 0.5f * pw - 1.f, y2 = pcy + 0.5f * ph - 1.f;
  x1 = fminf(fmaxf(x1, 0.f), IMG_W_F - 1.f);
  x2 = fminf(fmaxf(x2, 0.f), IMG_W_F - 1.f);
  y1 = fminf(fmaxf(y1, 0.f), IMG_H_F - 1.f);
  y2 = fminf(fmaxf(y2, 0.f), IMG_H_F - 1.f);

  scores[cls * N_BOX + n] = prob;
  ((float4*)boxes)[cls * N_BOX + n] = make_float4(x1, y1, x2, y2);
}

// ---------------- kernel 3: per-class bitonic sort (desc) in LDS ----------------
__global__ void __launch_bounds__(256) k_sort(const float* __restrict__ scores,
                                              const float* __restrict__ boxes,
                                              float* __restrict__ ssc,
                                              float* __restrict__ sbox) {
  __shared__ float key[NPAD];
  __shared__ int   sid[NPAD];
  int cls = blockIdx.x, tid = threadIdx.x;
  for (int e = 0; e < 4; ++e) {
    int s = e * 256 + tid;
    if (s < N_BOX) {
      float sc = scores[cls * N_BOX + s];
      key[s] = (sc > SCORE_TH) ? sc : NEG_INF_F;  // invalid sorted to the back
      sid[s] = s;
    } else {
      key[s] = -2e30f;  // pad slots last
      sid[s] = -1;
    }
  }
  __syncthreads();
  for (int k = 2; k <= NPAD; k <<= 1) {
    for (int j = k >> 1; j > 0; j >>= 1) {
      for (int e = 0; e < 4; ++e) {
        int i = e * 256 + tid;
        int p = i ^ j;
        if (p > i) {
          bool up = ((i & k) == 0);          // descending overall
          float ki = key[i], kp = key[p];
          bool sw = up ? (ki < kp) : (ki > kp);
          if (sw) {
            key[i] = kp; key[p] = ki;
            int t2 = sid[i]; sid[i] = sid[p]; sid[p] = t2;
          }
        }
      }
      __syncthreads();
    }
  }
  for (int e = 0; e < 4; ++e) {
    int s = e * 256 + tid;
    int id = sid[s];
    if (id >= 0) {
      ssc[cls * NPAD + s] = scores[cls * N_BOX + id];  // true score (even if invalid)
      ((float4*)sbox)[cls * NPAD + s] = ((const float4*)boxes)[cls * N_BOX + id];
    } else {
      ssc[cls * NPAD + s] = NEG_INF_F;
      // far-away degenerate pad box: zero overlap with any clipped real box
      ((float4*)sbox)[cls * NPAD + s] = make_float4(-1e4f, -1e4f, -1e4f, -1e4f);
    }
  }
}

// ---------------- kernel 4: suppression mask ----------------
// 8 blocks per class; TDM stages the class's 16KB sorted-box table into LDS once,
// then each wave32 owns one 16-row strip. WMMA computes pairwise area_i+area_j;
// IoU>0.5  <=>  3*inter > (area_i + area_j).
__global__ void __launch_bounds__(256) k_mask(const float* __restrict__ sbox,
                                              unsigned* __restrict__ maskp) {
  __shared__ float4 lbox[NPAD];          // 16 KB; single shared var -> LDS offset 0
  int tid  = threadIdx.x;
  int lane = tid & 31;
  int wv   = tid >> 5;                   // wave id 0..7
  int cls  = blockIdx.x >> 3;
  int sub  = blockIdx.x & 7;
  int lane16 = lane & 15;
  bool lo = lane < 16;
  const float4* SB = (const float4*)sbox + (size_t)cls * NPAD;

  if (wv == 0) {
    // Tensor Data Mover: 1-D tile, 2048 x 8-byte units = 16 KB, global -> LDS@0.
    unsigned long long ga = (unsigned long long)(uintptr_t)SB;
    uv4 g0;
    g0[0] = 1u;                                           // count=1, user desc
    g0[1] = 0u;                                           // lds_addr = 0
    g0[2] = (unsigned)ga;                                 // global_addr[31:0]
    g0[3] = (unsigned)((ga >> 32) & 0x1FFFFFFu) | (2u << 30);  // addr[56:32] | type=2
    uv8 g1;
    g1[0] = (3u << 16);                                   // data_size=8B, mask=0
    g1[1] = (2048u & 0xFFFFu) << 16;                      // tensor_dim0[15:0]
    g1[2] = (2048u >> 16) | (1u << 16);                   // tensor_dim0[31:16] | tensor_dim1=1
    g1[3] = (2048u << 16);                                // tile_dim0 = 2048
    g1[4] = 0u;                                           // tile_dim1/2 unused
    g1[5] = 2048u;                                        // tensor_dim0_stride lo
    g1[6] = 0u;
    g1[7] = 0u;
    // NOTE: lbox is passed as an operand so the LDS buffer ESCAPES into the asm;
    // without this, alias analysis proves the asm can't touch the non-escaping
    // __shared__ array and deletes all subsequent ds_loads from it.
    asm volatile("tensor_load_to_lds %0, %1"
                 :
                 : "s"(g0), "s"(g1), "v"((void*)lbox)
                 : "memory");
    __builtin_amdgcn_s_wait_tensorcnt(0);
  }
  __syncthreads();

  int tr = sub * 8 + wv;                 // 0..63; strip 63 idles (wave-uniform guard)
  if (tr < NTILE) {
    // A operand (16x4 f32): row M -> [area(M), 1, 0, 0]
    float4 ra = lbox[tr * 16 + lane16];
    float areaA = (ra.z - ra.x + 1.f) * (ra.w - ra.y + 1.f);
    v2f va;
    va.x = lo ? areaA : 0.f;
    va.y = lo ? 1.f   : 0.f;

    // per-lane row boxes: lanes<16 -> rows M=0..7, lanes>=16 -> rows M=8..15
    int rowbase = tr * 16 + (lo ? 0 : 8);
    float4 rb[8];
    for (int r = 0; r < 8; ++r) rb[r] = lbox[rowbase + r];

    unsigned pairW = 0;
    for (int tc = 0; tc < NTILE; ++tc) {
      float4 cb = lbox[tc * 16 + lane16];          // column box from LDS (ds_load_b128)
      float areaB = (cb.z - cb.x + 1.f) * (cb.w - cb.y + 1.f);
      // B operand (4x16 f32): col N -> [1, area(N), 0, 0]^T
      v2f vb;
      vb.x = lo ? 1.f   : 0.f;
      vb.y = lo ? areaB : 0.f;
      v8f cz = {0.f, 0.f, 0.f, 0.f, 0.f, 0.f, 0.f, 0.f};
      // D[i][j] = area_i + area_j  (EXEC all ones inside this wave-uniform region)
      v8f d = __builtin_amdgcn_wmma_f32_16x16x4_f32(false, va, false, vb,
                                                    (short)0, cz, false, false);
      unsigned rowbits = 0;
      for (int r = 0; r < 8; ++r) {
        float iw = fminf(rb[r].z, cb.z) - fmaxf(rb[r].x, cb.x) + 1.f;
        float ih = fminf(rb[r].w, cb.w) - fmaxf(rb[r].y, cb.y) + 1.f;
        float inter = fmaxf(iw, 0.f) * fmaxf(ih, 0.f);
        bool sup = (3.f * inter > d[r]);           // strict, matches iou > 0.5
        unsigned m = __builtin_amdgcn_ballot_w32(sup);
        // D layout: lanes0-15 vgpr r -> (M=r, N=lane); lanes16-31 -> (M=8+r, N=lane-16)
        if (lane == r)     rowbits = m & 0xFFFFu;  // row tr*16 + r
        if (lane == 8 + r) rowbits = m >> 16;      // row tr*16 + 8 + r
      }
      if ((tc & 1) == 0) pairW = rowbits;
      else               pairW |= rowbits << 16;
      if ((tc & 1) == 1 || tc == NTILE - 1) {
        if (lane < 16)
          maskp[((size_t)cls * NPAD + tr * 16 + lane) * 32 + (tc >> 1)] = pairW;
      }
    }
  }
}

// ---------------- kernel 5: greedy scan, one wave per class ----------------
__global__ void __launch_bounds__(32) k_scan(const float* __restrict__ ssc,
                                             const unsigned* __restrict__ maskp,
                                             unsigned* __restrict__ keepw) {
  int cls = blockIdx.x;
  int lane = threadIdx.x;           // lane w owns removal word w
  unsigned remv = 0u, keep = 0u;
  for (int i = 0; i < N_BOX; ++i) {
    int w = i >> 5, bit = i & 31;
    float sc = ssc[(size_t)cls * NPAD + i];                 // broadcast load
    unsigned rw = __builtin_amdgcn_readlane(remv, w);       // uniform lane select
    bool alive = (sc > SCORE_TH) && !((rw >> bit) & 1u);
    if (alive) {
      if (lane == w) keep |= (1u << bit);
      remv |= maskp[((size_t)cls * NPAD + i) * 32 + lane];  // coalesced 32-word row
    }
  }
  keepw[cls * 32 + lane] = keep;
}

// ---------------- kernel 6a: build flat mutable score array ----------------
__global__ void k_flat(const float* __restrict__ ssc,
                       const unsigned* __restrict__ keepw,
                       float* __restrict__ flat) {
  int t = blockIdx.x * blockDim.x + threadIdx.x;
  if (t >= NCLS * N_BOX) return;
  int cls = t / N_BOX, k = t % N_BOX;
  bool kp = (keepw[cls * 32 + (k >> 5)] >> (k & 31)) & 1u;
  // flat index t == b*80000 + c1*1000 + k  (same flatten order as reference)
  flat[t] = kp ? ssc[(size_t)cls * NPAD + k] : NEG_INF_F;
}

// ---------------- kernel 6b: per-image top-100 (argmax rounds, low-index ties) ----------------
__global__ void __launch_bounds__(256) k_topk(const float* __restrict__ sbox,
                                              float* __restrict__ flat,
                                              float* __restrict__ out) {
  __shared__ float rs[256];
  __shared__ int   ri[256];
  int b = blockIdx.x, tid = threadIdx.x;
  float* fb = flat + (size_t)b * C1 * N_BOX;
  for (int round = 0; round < DETS; ++round) {
    float bs = DEAD_F;
    int bi = 0x7fffffff;
    for (int f = tid; f < C1 * N_BOX; f += 256) {
      float v = fb[f];
      if (v > bs || (v == bs && f < bi)) { bs = v; bi = f; }
    }
    rs[tid] = bs; ri[tid] = bi;
    __syncthreads();
    for (int s = 128; s > 0; s >>= 1) {
      if (tid < s) {
        if (rs[tid + s] > rs[tid] ||
            (rs[tid + s] == rs[tid] && ri[tid + s] < ri[tid])) {
          rs[tid] = rs[tid + s];
          ri[tid] = ri[tid + s];
        }
      }
      __syncthreads();
    }
    if (tid == 0) {
      int f = ri[0];
      float s0 = rs[0];
      bool vOK = s0 > -5e29f;       // matches top_s > NEG_INF*0.5
      int c1 = f / N_BOX, k = f % N_BOX;
      float4 bx = ((const float4*)sbox)[((size_t)b * C1 + c1) * NPAD + k];
      float* ob = out + (b * DETS + round) * 4;
      ob[0] = bx.x; ob[1] = bx.y; ob[2] = bx.z; ob[3] = bx.w;
      out[800  + b * DETS + round] = vOK ? s0 : 0.f;
      out[1000 + b * DETS + round] = (float)(c1 + 1);
      out[1200 + b * DETS + round] = vOK ? 1.f : 0.f;
      fb[f] = DEAD_F;               // consume
    }
    __syncthreads();
  }
}

extern "C" void kernel_launch(void* const* d_in, const int* in_sizes, int n_in,
                              void* d_out, int out_size, void* d_ws, size_t ws_size,
                              hipStream_t stream) {
  (void)in_sizes; (void)n_in; (void)out_size; (void)ws_size;
  const float* logits = (const float*)d_in[0];   // [2000,81]
  const float* boxreg = (const float*)d_in[1];   // [2000,324]
  const float* props  = (const float*)d_in[2];   // [2,1000,4]
  float* out = (float*)d_out;                    // 800 boxes + 200 scores + 200 labels + 200 valid

  char* p = (char*)d_ws;
  auto carve = [&](size_t bytes) -> char* {
    char* r = p;
    p += (bytes + 255) & ~(size_t)255;
    return r;
  };
  float*    rowmax = (float*)carve(2000 * sizeof(float));
  float*    rowsum = (float*)carve(2000 * sizeof(float));
  float*    scores = (float*)carve((size_t)NCLS * N_BOX * sizeof(float));
  float*    boxes  = (float*)carve((size_t)NCLS * N_BOX * 4 * sizeof(float));
  float*    ssc    = (float*)carve((size_t)NCLS * NPAD * sizeof(float));
  float*    sbox   = (float*)carve((size_t)NCLS * NPAD * 4 * sizeof(float));
  unsigned* keepw  = (unsigned*)carve((size_t)NCLS * 32 * sizeof(unsigned));
  float*    flat   = (float*)carve((size_t)B_IMG * C1 * N_BOX * sizeof(float));
  unsigned* maskp  = (unsigned*)carve((size_t)NCLS * NPAD * 32 * sizeof(unsigned));

  k_softmax_stats<<<8, 256, 0, stream>>>(logits, rowmax, rowsum);
  k_decode<<<(NCLS * N_BOX + 255) / 256, 256, 0, stream>>>(logits, boxreg, props,
                                                           rowmax, rowsum, scores, boxes);
  k_sort<<<NCLS, 256, 0, stream>>>(scores, boxes, ssc, sbox);
  k_mask<<<NCLS * 8, 256, 0, stream>>>(sbox, maskp);   // 8 strip-blocks per class
  k_scan<<<NCLS, 32, 0, stream>>>(ssc, maskp, keepw);
  k_flat<<<(NCLS * N_BOX + 255) / 256, 256, 0, stream>>>(ssc, keepw, flat);
  k_topk<<<B_IMG, 256, 0, stream>>>(sbox, flat, out);
}